// CTCLayer_8203387535679
// MI455X (gfx1250) — compile-verified
//
#include <hip/hip_runtime.h>
#include <stdint.h>
#include <math.h>

// Problem constants (from reference): B=256, T=512, C=128, L=96
#define BB 256
#define TT 512
#define CC 128
#define LL 96
#define SS 193              // 2*L + 1 real states
#define KK 7                // state slots per lane: 7*32 = 224 >= 193 (padded, branchless)
#define NEGV (-1.7014118e38f)   // finfo(f32).min / 2, matches reference sentinel

// pairwise logaddexp (used once, at the end)
__device__ __forceinline__ float lae(float a, float b) {
    float mx = fmaxf(a, b);
    float mn = fminf(a, b);
    return mx + log1pf(expf(mn - mx));
}

// 3-way log-sum-exp; z == NEGV contributes exactly 0 (expf underflows),
// making the CTC skip-transition branchless.
__device__ __forceinline__ float lae3(float x, float y, float z) {
    float mx = fmaxf(fmaxf(x, y), z);
    float s = expf(x - mx) + expf(y - mx) + expf(z - mx);
    return mx + logf(s);
}

// Async global->LDS: one 512B logit row as 32 lanes x 16B transfers (ASYNCcnt).
__device__ __forceinline__ void issue_row_async(const float* gbase, int t, uint32_t lds_lane) {
    uint32_t voff = (uint32_t)t * (CC * 4) + (uint32_t)threadIdx.x * 16u;
    asm volatile("global_load_async_to_lds_b128 %0, %1, %2"
                 :: "v"(lds_lane), "v"(voff), "s"(gbase)
                 : "memory");
}

// Wave32 row log-sum-exp (off the alpha critical path; feeds scalar accumulator only).
__device__ __forceinline__ float row_lse(const float* row) {
    int lane = threadIdx.x;
    float4 v = *(const float4*)(row + lane * 4);
    float m = fmaxf(fmaxf(v.x, v.y), fmaxf(v.z, v.w));
#pragma unroll
    for (int off = 16; off >= 1; off >>= 1)
        m = fmaxf(m, __shfl_xor(m, off, 32));
    float s = expf(v.x - m) + expf(v.y - m) + expf(v.z - m) + expf(v.w - m);
#pragma unroll
    for (int off = 16; off >= 1; off >>= 1)
        s += __shfl_xor(s, off, 32);
    return m + logf(s);
}

__global__ __launch_bounds__(32)
void ctc_fwd_kernel(const float* __restrict__ y_pred,
                    const int*   __restrict__ y_true,
                    float*       __restrict__ out) {
    __shared__ __align__(16) float rowbuf[2][CC];   // only LDS: double-buffered rows

    const int b    = blockIdx.x;
    const int lane = threadIdx.x;
    const float* gbase = y_pred + (size_t)b * TT * CC;
    const int*   lab   = y_true + b * LL;

    // LDS byte addresses (low 32 bits of generic LDS pointer == DS offset)
    const uint32_t lds0 = (uint32_t)(uintptr_t)(&rowbuf[0][0]) + (uint32_t)lane * 16u;
    const uint32_t lds1 = (uint32_t)(uintptr_t)(&rowbuf[1][0]) + (uint32_t)lane * 16u;

    // prefetch rows 0 and 1 immediately; overlap with label setup below
    issue_row_async(gbase, 0, lds0);
    issue_row_async(gbase, 1, lds1);

    // ---- per-lane extended labels + skip flags (registers, loop-invariant) ----
    int  extR[KK];
    bool skipR[KK];
#pragma unroll
    for (int k = 0; k < KK; ++k) {
        int s = lane + 32 * k;
        int e = 0; bool sk = false;
        if (s & 1) {
            int li = s >> 1; if (li > LL - 1) li = LL - 1;   // clamp pad states
            e = lab[li];
            int li1 = (li > 0) ? (li - 1) : 0;               // no OOB speculation
            sk = (s >= 3) && (s < SS) && (e != lab[li1]);
        }
        extR[k] = e; skipR[k] = sk;
    }

    // ---- t = 0 : recursion runs on RAW logits; normalization folded into cum ----
    asm volatile("s_wait_asynccnt 0x1" ::: "memory");   // in-order: row 0 resident
    double cum = (double)row_lse(rowbuf[0]);            // sum of per-step LSEs (f64, off-path)
    float a[KK];
#pragma unroll
    for (int k = 0; k < KK; ++k) {
        int s = lane + 32 * k;
        a[k] = (s < 2) ? rowbuf[0][extR[k]] : NEGV;     // only states 0,1 reachable at t=0
    }

    // ---- forward recursion, alpha fully register-resident ----
    for (int t = 1; t < TT; ++t) {
        const int bi   = t & 1;
        const int trow = (t + 1 < TT) ? (t + 1) : (TT - 1);   // last iter: dead-buffer reload
        issue_row_async(gbase, trow, ((t + 1) & 1) ? lds1 : lds0);
        asm volatile("s_wait_asynccnt 0x1" ::: "memory");     // row t resident
        const float* row = rowbuf[bi];
        cum += (double)row_lse(row);                          // independent of alpha chain

        // state shifts s-1 / s-2 via wave32 shuffles (old alpha values)
        float p1[KK], p2[KK];
#pragma unroll
        for (int k = 0; k < KK; ++k) {
            float prevk = (k == 0) ? NEGV : a[k - 1];
            float w31 = __shfl(prevk, 31, 32);                // state s-1 for lane 0
            float wx  = __shfl(prevk, (30 + lane) & 31, 32);  // state s-2 for lanes 0,1
            float u1  = __shfl_up(a[k], 1, 32);
            float u2  = __shfl_up(a[k], 2, 32);
            p1[k] = (lane == 0) ? w31 : u1;
            p2[k] = (lane < 2)  ? wx  : u2;
        }
#pragma unroll
        for (int k = 0; k < KK; ++k) {
            float p2e = skipR[k] ? p2[k] : NEGV;   // branchless skip transition
            a[k] = lae3(a[k], p1[k], p2e) + row[extR[k]];
        }
    }

    // ends in last blank (s=192: lane0,k=6) or last label (s=191: lane31,k=5)
    float v191 = __shfl(a[5], 31, 32);
    float v192 = __shfl(a[6], 0, 32);
    if (lane == 0)
        out[b] = (float)(cum - (double)lae(v192, v191));   // loss = Σ lse − log p
}

extern "C" void kernel_launch(void* const* d_in, const int* in_sizes, int n_in,
                              void* d_out, int out_size, void* d_ws, size_t ws_size,
                              hipStream_t stream) {
    const float* y_pred = (const float*)d_in[0];   // [B,T,C] f32
    const int*   y_true = (const int*)d_in[1];     // [B,L]   i32
    float*       out    = (float*)d_out;           // [B]     f32
    (void)in_sizes; (void)n_in; (void)out_size; (void)d_ws; (void)ws_size;
    ctc_fwd_kernel<<<dim3(BB), dim3(32), 0, stream>>>(y_pred, y_true, out);
}